// ARP_62251255988732
// MI455X (gfx1250) — compile-verified
//
#include <hip/hip_runtime.h>
#include <stdint.h>

// ---------------------------------------------------------------------------
// ARP torus AR(3) log-prob, MI455X (gfx1250).
// Memory-bound: 12.6 MB in -> 128 B out. Strategy:
//   kernel 1: one 256-thread block per (mc, sample) row (512 blocks).
//     - async global->LDS b128 copy of the 24 KB row (ASYNCcnt path)
//     - dx = wrap(diff) into LDS
//     - u = (dy-c)/s via V_WMMA_F32_16X16X4_F32: 4-tap FIR as 16x16x4 matmul
//       with scale folded into B and -c/s folded into the C accumulator
//     - accumulate u^2, block-reduce, write partial to ws
//   kernel 2: 1 wave folds 16 partials per mc + analytic winding constant
//       (sum over 11 winding offsets collapses to -5.5u^2 + const).
// ---------------------------------------------------------------------------

#define PI_F          3.14159265358979323846f
#define TWO_PI_F      6.28318530717958647692f
#define INV_TWO_PI_F  0.15915494309189533577f

typedef __attribute__((ext_vector_type(2))) float v2f;
typedef __attribute__((ext_vector_type(8))) float v8f;

#define T_LEN      2048
#define D_DIM      3
#define ROW_FLOATS (T_LEN * D_DIM)     // 6144 floats = 24576 B per (mc,sample) row
#define DX_LEN     (T_LEN - 1)         // 2047 increments
#define TP_LEN     (DX_LEN - 3)        // 2044 residuals
#define NFULL      127                 // full 16-wide t' blocks (127*16 = 2032)
#define TAIL_START (NFULL * 16)        // 2032
#define TAIL_CNT   (TP_LEN - TAIL_START) // 12

#if __has_builtin(__builtin_amdgcn_wmma_f32_16x16x4_f32)
#define HAVE_WMMA_F32_16X16X4 1
#else
#define HAVE_WMMA_F32_16X16X4 0
#endif

__device__ __forceinline__ float torus_wrap(float x) {
  // jnp.mod(x + pi, 2pi) - pi  (mod result in [0, 2pi))
  float y = x + PI_F;
  y -= TWO_PI_F * floorf(y * INV_TWO_PI_F);
  return y - PI_F;
}

__global__ __launch_bounds__(256)
void arp_row_kernel(const float* __restrict__ g,
                    const float* __restrict__ ar_phi,
                    const float* __restrict__ ar_eta,
                    const float* __restrict__ ar_c,
                    float* __restrict__ ws) {
  __shared__ float sh_g[ROW_FLOATS];          // raw angles of this row
  __shared__ float sh_dx[DX_LEN * D_DIM];     // wrapped increments
  __shared__ float sh_red[256];               // block reduction scratch

  const int tid  = threadIdx.x;
  const int lane = tid & 31;
  const int wv   = tid >> 5;
  const int row  = blockIdx.x;                // 0..511 == mc*16 + sample
  const float* grow = g + (size_t)row * ROW_FLOATS;

  // ---- uniform small parameters (scalar loads) ----
  float ph[D_DIM][3], cc[D_DIM], inv_s[D_DIM];
  #pragma unroll
  for (int d = 0; d < D_DIM; ++d) {
    #pragma unroll
    for (int p = 0; p < 3; ++p) ph[d][p] = ar_phi[d * 3 + p];
    cc[d]    = ar_c[d];
    inv_s[d] = 1.0f / fabsf(ar_eta[d]);
  }

  // ---- phase 1: async global -> LDS copy (b128 lanes, ASYNCcnt) ----
  // 6144 floats = 1536 x b128; 256 threads x 6 iterations.
  #pragma unroll
  for (int it = 0; it < 6; ++it) {
    int i4 = it * 256 + tid;                              // float4 index
    unsigned lds_off = (unsigned)(uintptr_t)(const void*)(&sh_g[i4 * 4]);
    unsigned long long gaddr = (unsigned long long)(uintptr_t)(const void*)(grow + i4 * 4);
    asm volatile("global_load_async_to_lds_b128 %0, %1, off"
                 :: "v"(lds_off), "v"(gaddr) : "memory");
  }
  asm volatile("s_wait_asynccnt 0" ::: "memory");
  __syncthreads();

  // ---- phase 2a: dx = wrap(g[t+1] - g[t]) ----
  for (int i = tid; i < DX_LEN * D_DIM; i += 256) {
    sh_dx[i] = torus_wrap(sh_g[i + D_DIM] - sh_g[i]);
  }
  __syncthreads();

  // ---- phase 2b: u = (dy-c)/s via WMMA f32 16x16x4, accumulate u^2 ----
  // A[m][k] = dx(t0+m+k)  (lane L: M=L%16; VGPR0/1 hold K = 2*(L/16), 2*(L/16)+1)
  // B[k][n] = w_k/s_d for all n, C[m][n] = -c_d/s_d
  //   -> D[m][n] = (dy(t0+m) - c_d)/s_d, replicated over the 16 columns.
  const int h = lane >> 4;         // which K-half this lane holds
  const int m = lane & 15;
  float accW = 0.0f;               // 16x overcounted (replicated columns)
  float accT = 0.0f;               // tail, counted once

  #pragma unroll
  for (int d = 0; d < D_DIM; ++d) {
    const float isd = inv_s[d];
    // FIR taps (dy = dx[t+3] - phi0*dx[t+2] - phi1*dx[t+1] - phi2*dx[t]),
    // pre-scaled by 1/s.
    const float w0 = -ph[d][2] * isd;
    const float w1 = -ph[d][1] * isd;
    const float w2 = -ph[d][0] * isd;
    const float w3 = isd;
    v2f bb;  bb.x = h ? w2 : w0;  bb.y = h ? w3 : w1;     // hoisted per d
    const float cbias = -cc[d] * isd;
    v8f cfrag;
    #pragma unroll
    for (int r = 0; r < 8; ++r) cfrag[r] = cbias;
    const float* dxd = sh_dx + d;                          // stride-3 view

    for (int b = wv; b < NFULL; b += 8) {                  // wave-uniform trips
      const int tb = (b << 4) + m + (h << 1);
#if HAVE_WMMA_F32_16X16X4
      v2f a;  a.x = dxd[tb * D_DIM];  a.y = dxd[(tb + 1) * D_DIM];
      v8f u = __builtin_amdgcn_wmma_f32_16x16x4_f32(
          /*neg_a=*/false, a, /*neg_b=*/false, bb,
          /*c_mod=*/(short)0, cfrag, /*reuse_a=*/false, /*reuse_b=*/false);
      #pragma unroll
      for (int r = 0; r < 8; ++r) accW += u[r] * u[r];     // v_fmac chain
#else
      // VALU fallback: each m appears in 2 lanes; weight by 8 to keep the
      // same 16x overcount convention as the WMMA path.
      const int tp = (b << 4) + m;
      float u = cbias + w3 * dxd[(tp + 3) * D_DIM] + w2 * dxd[(tp + 2) * D_DIM]
                      + w1 * dxd[(tp + 1) * D_DIM] + w0 * dxd[tp * D_DIM];
      accW += 8.0f * u * u;
#endif
    }
  }

  // ---- tail t' = 2032..2043, scalar ----
  if (tid < TAIL_CNT * D_DIM) {
    int tp = TAIL_START + tid / D_DIM;
    int d  = tid % D_DIM;
    float dy = sh_dx[(tp + 3) * D_DIM + d]
             - (ph[d][0] * sh_dx[(tp + 2) * D_DIM + d]
              + ph[d][1] * sh_dx[(tp + 1) * D_DIM + d]
              + ph[d][2] * sh_dx[tp * D_DIM + d]);
    float u = (dy - cc[d]) * inv_s[d];
    accT += u * u;
  }

  // ---- deterministic block reduction ----
  sh_red[tid] = -5.5f * (accW * (1.0f / 16.0f) + accT);
  __syncthreads();
  #pragma unroll
  for (int s = 128; s > 0; s >>= 1) {
    if (tid < s) sh_red[tid] += sh_red[tid + s];
    __syncthreads();
  }
  if (tid == 0) ws[row] = sh_red[0];
}

__global__ void arp_finalize_kernel(const float* __restrict__ ar_eta,
                                    const float* __restrict__ ws,
                                    float* __restrict__ out) {
  int m = threadIdx.x;
  if (m >= 32) return;
  float s = 0.0f;
  #pragma unroll
  for (int i = 0; i < 16; ++i) s += ws[m * 16 + i];
  // analytic winding-sum constant per (t',d) term:
  //   -220*pi^2/s_d^2 - 11*log s_d - 5.5*log(2*pi)
  float csum = 0.0f;
  #pragma unroll
  for (int d = 0; d < D_DIM; ++d) {
    float sd = fabsf(ar_eta[d]);
    csum += -220.0f * (PI_F * PI_F) / (sd * sd)
            - 11.0f * logf(sd)
            - 5.5f * logf(TWO_PI_F);
  }
  out[m] = s + 16.0f * (float)TP_LEN * csum;
}

extern "C" void kernel_launch(void* const* d_in, const int* in_sizes, int n_in,
                              void* d_out, int out_size, void* d_ws, size_t ws_size,
                              hipStream_t stream) {
  const float* g      = (const float*)d_in[0];
  const float* ar_phi = (const float*)d_in[1];
  const float* ar_eta = (const float*)d_in[2];
  const float* ar_c   = (const float*)d_in[3];
  float* out = (float*)d_out;
  float* ws  = (float*)d_ws;          // 512 floats of partials

  arp_row_kernel<<<512, 256, 0, stream>>>(g, ar_phi, ar_eta, ar_c, ws);
  arp_finalize_kernel<<<1, 32, 0, stream>>>(ar_eta, ws, out);
}